// RBF_1202590843708
// MI455X (gfx1250) — compile-verified
//
#include <hip/hip_runtime.h>

// RBF kernel: out[r,m,n] = os^2 * exp(-0.5 * max(sqdist, 0))
// sqdist computed entirely inside the matrix pipe via an augmented dot product,
// with the constant s = -0.5*log2(e) pre-folded into the B operand:
//   u = [a_0..a_7, aa, 1, 0, 0]                       (A rows, a = x1 * inv_ls)
//   v = [log2e*b_0..log2e*b_7, s, s*bb, 0, 0]         (B cols, b = x2 * inv_ls)
//   dot(u,v) = s * (aa + bb - 2 a.b) = s * sqdist     (s < 0)
// K=12 -> three chained V_WMMA_F32_16X16X4_F32. Since s < 0:
//   -0.5*log2e*max(sq,0) = min(s*sq, 0), then v_exp_f32 (exp2) + sigma^2 mul.
// Each wave computes a 16x64 strip (4 n-subtiles) to amortize A-side work,
// tile decode and address math; all loads/stores use immediate offsets.
// Store-bandwidth bound: 268 MB output @ 23.3 TB/s ~= 11.5 us floor.

typedef __attribute__((ext_vector_type(2))) float v2f;
typedef __attribute__((ext_vector_type(8))) float v8f;

#define R_  4
#define N1_ 4096
#define N2_ 4096
#define D_  8

__global__ __launch_bounds__(256) void rbf_wmma_kernel(
    const float* __restrict__ x1,   // [R, N1, D]
    const float* __restrict__ x2,   // [R, N2, D]
    const float* __restrict__ ls,   // [D, 1]
    const float* __restrict__ os,   // [1]
    float* __restrict__ out)        // [R, N1, N2]
{
    const int lane = threadIdx.x & 31;
    const int wave = threadIdx.x >> 5;
    // Wave work unit: 16 (M) x 64 (N) strip. 4 * 256 * 64 = 65536 units.
    const unsigned u = blockIdx.x * 8u + (unsigned)wave;
    const int r  = (int)(u >> 14);        // [0,4)
    const int mt = (int)((u >> 6) & 255); // M tile   [0,256)
    const int ng = (int)(u & 63);         // N group  [0,64)
    const int m0 = mt * 16;
    const int n0 = ng * 64;

    const int row  = lane & 15;   // M row (A) / N col (B, C)
    const int hi   = lane >> 4;   // lane half selects K pair {0,1} vs {2,3}
    const int koff = hi * 2;

    const float LOG2E = 1.4426950408889634f;   // -2*s
    const float S     = -0.5f * LOG2E;         // s

    // Per-lane inverse lengthscales (lane-dependent addresses, no selects).
    v2f il0 = { 1.0f / ls[koff + 0], 1.0f / ls[koff + 1] };
    v2f il1 = { 1.0f / ls[4 + koff + 0], 1.0f / ls[4 + koff + 1] };
    const float osv = os[0];
    const float os2 = osv * osv;

    // ---- A side (computed once per strip) ----
    const float* arow = x1 + ((size_t)r * N1_ + (size_t)(m0 + row)) * D_;
    v2f a0 = *(const v2f*)(arow + koff)     * il0;
    v2f a1 = *(const v2f*)(arow + 4 + koff) * il1;

    float aa_p = a0.x * a0.x + a0.y * a0.y + a1.x * a1.x + a1.y * a1.y;
    // Complementary K half lives in lane^16: ds_swizzle SWAPX16 (xor 0x10).
    float aa = aa_p + __int_as_float(
        __builtin_amdgcn_ds_swizzle(__float_as_int(aa_p), 0x401f));

    const float A1 = hi ? 0.0f : 1.0f;  // masks augmented K=10,11 lanes to zero
    const float Sc = A1 * S;
    v2f a2 = { A1 * aa, A1 };           // A aug chunk: {aa,1} / {0,0}

    // One 64-bit store base for the whole strip; everything else is imm offset.
    float* optr = out + ((size_t)r * N1_ + (size_t)(m0 + hi * 8)) * (size_t)N2_
                      + (size_t)(n0 + row);
    const float* brow = x2 + ((size_t)r * N2_ + (size_t)(n0 + row)) * D_;

#pragma unroll
    for (int t = 0; t < 4; ++t) {
        // ---- B side for n-subtile t (rows n0 + t*16 + row) ----
        const float* br = brow + t * 16 * D_;
        v2f b0 = *(const v2f*)(br + koff)     * il0;
        v2f b1 = *(const v2f*)(br + 4 + koff) * il1;

        float bb_p = b0.x * b0.x + b0.y * b0.y + b1.x * b1.x + b1.y * b1.y;
        float bb = bb_p + __int_as_float(
            __builtin_amdgcn_ds_swizzle(__float_as_int(bb_p), 0x401f));

        v2f b2 = { Sc, Sc * bb };       // B aug chunk: {s, s*bb} / {0,0}
        b0 *= LOG2E;                     // -2*s folded into first 8 K
        b1 *= LOG2E;

        v8f c = {};
        c = __builtin_amdgcn_wmma_f32_16x16x4_f32(false, a0, false, b0,
                                                  (short)0, c, false, false);
        c = __builtin_amdgcn_wmma_f32_16x16x4_f32(false, a1, false, b1,
                                                  (short)0, c, false, false);
        c = __builtin_amdgcn_wmma_f32_16x16x4_f32(false, a2, false, b2,
                                                  (short)0, c, false, false);
        // c[v] = s * sqdist at M = v + 8*hi, N = n0 + t*16 + row.

#pragma unroll
        for (int v = 0; v < 8; ++v) {
            float q = c[v];
            float e = (q < 0.0f) ? q : 0.0f;       // min(s*sq, 0) == s*max(sq,0)
            optr[(size_t)v * N2_ + t * 16] = os2 * __builtin_amdgcn_exp2f(e);
        }
    }
}

extern "C" void kernel_launch(void* const* d_in, const int* in_sizes, int n_in,
                              void* d_out, int out_size, void* d_ws, size_t ws_size,
                              hipStream_t stream) {
    const float* x1 = (const float*)d_in[0];
    const float* x2 = (const float*)d_in[1];
    const float* ls = (const float*)d_in[2];
    const float* os = (const float*)d_in[3];
    float* out = (float*)d_out;

    const int units  = R_ * (N1_ / 16) * (N2_ / 64);  // 65536 16x64 strips
    const int blocks = units / 8;                     // 8 waves per 256-thread block
    rbf_wmma_kernel<<<blocks, 256, 0, stream>>>(x1, x2, ls, os, out);
}